// BinaryTreeRNN_22342419874355
// MI455X (gfx1250) — compile-verified
//
#include <hip/hip_runtime.h>
#include <math.h>

typedef __attribute__((ext_vector_type(2))) float v2f;
typedef __attribute__((ext_vector_type(8))) float v8f;

#define LDS_STRIDE 10   // 40B row stride: 8B aligned, conflict-free across 32 lanes

__global__ __launch_bounds__(256) void
binary_tree_rnn_kernel(const float* __restrict__ x,
                       const float* __restrict__ W_leaf,
                       const float* __restrict__ b_leaf,
                       const float* __restrict__ w0, const float* __restrict__ w1,
                       const float* __restrict__ w2, const float* __restrict__ b0,
                       const float* __restrict__ b1, const float* __restrict__ b2,
                       const float* __restrict__ om0, const float* __restrict__ om1,
                       const float* __restrict__ om2,
                       float* __restrict__ out, int N)
{
    __shared__ float lds_h[8][32 * LDS_STRIDE];   // per-wave private slabs (10 KB)

    const int tid    = threadIdx.x;
    const int lane   = tid & 31;
    const int waveId = tid >> 5;
    const int n      = lane & 15;        // tile column (A-row / B-col index)
    const int khalf  = (lane >> 4) & 1;  // 0 -> K {0,1}+4j ; 1 -> K {2,3}+4j

    // ---- B fragments: B[k][n] = W_leaf[n][k] for n<8, zero-padded n>=8 ----
    v2f bfrag[4];
#pragma unroll
    for (int j = 0; j < 4; ++j) {
        const int k0 = 4 * j + 2 * khalf;
        float v0 = 0.f, v1 = 0.f;
        if (n < 8) {
            v0 = W_leaf[n * 16 + k0];
            v1 = W_leaf[n * 16 + k0 + 1];
        }
        bfrag[j] = (v2f){v0, v1};
    }
    const float cbias = (n < 8) ? b_leaf[n] : 0.f;

    // ---- per-node coefficients: val = A*s + B*(l*r) + C*sin(s) + D ----
    // nodes 0..3 = level2 (om2,w2,b2), 4..5 = level1, 6 = level0
    float cA[7], cB[7], cC[7], cD[7];
    {
        const float* oms[3] = { om2, om1, om0 };
        const float* ws[3]  = { w2,  w1,  w0  };
        const float* bs[3]  = { b2,  b1,  b0  };
        const int cnt[3]    = { 4, 2, 1 };
        int idx = 0;
        for (int L = 0; L < 3; ++L) {
            for (int i = 0; i < cnt[L]; ++i) {
                const float o0 = oms[L][i*4+0], o1 = oms[L][i*4+1];
                const float o2 = oms[L][i*4+2], o3 = oms[L][i*4+3];
                const float mx = fmaxf(fmaxf(o0, o1), fmaxf(o2, o3));
                const float e0 = __expf(o0 - mx), e1 = __expf(o1 - mx);
                const float e2 = __expf(o2 - mx), e3 = __expf(o3 - mx);
                const float inv = 1.f / (e0 + e1 + e2 + e3);
                const float w = ws[L][i];
                cA[idx] = w * (e0 + e3) * inv;   // sm0*s + sm3*s
                cB[idx] = w * e1 * inv;          // sm1*(l*r)
                cC[idx] = w * e2 * inv;          // sm2*sin(s)
                cD[idx] = bs[L][i];
                ++idx;
            }
        }
    }

    float* myLds = lds_h[waveId];

    const long long tilesTotal  = ((long long)N + 31) / 32;          // 32 rows / wave / iter
    const long long wavesGlobal = (long long)gridDim.x * (blockDim.x >> 5);
    const long long gwave       = (long long)blockIdx.x * (blockDim.x >> 5) + waveId;

    for (long long t = gwave; t < tilesTotal; t += wavesGlobal) {
        const long long rowBase = t * 32;

        v8f acc[2];
#pragma unroll
        for (int half = 0; half < 2; ++half) {
            long long r = rowBase + half * 16 + n;       // A-matrix row for this lane
            if (r > (long long)N - 1) r = (long long)N - 1;  // tail clamp (EXEC stays full)
            const float* p = x + r * 16 + 2 * khalf;
            const v2f a0 = *(const v2f*)(p + 0);
            const v2f a1 = *(const v2f*)(p + 4);
            const v2f a2 = *(const v2f*)(p + 8);
            const v2f a3 = *(const v2f*)(p + 12);
            v8f c = { cbias, cbias, cbias, cbias, cbias, cbias, cbias, cbias };
            c = __builtin_amdgcn_wmma_f32_16x16x4_f32(false, a0, false, bfrag[0], (short)0, c, false, false);
            c = __builtin_amdgcn_wmma_f32_16x16x4_f32(false, a1, false, bfrag[1], (short)0, c, false, false);
            c = __builtin_amdgcn_wmma_f32_16x16x4_f32(false, a2, false, bfrag[2], (short)0, c, false, false);
            c = __builtin_amdgcn_wmma_f32_16x16x4_f32(false, a3, false, bfrag[3], (short)0, c, false, false);
            acc[half] = c;
        }

        // ---- scatter D tiles to LDS: lane owns column n, rows mBase..mBase+7 ----
        const int mBase = khalf * 8;
        if (n < 8) {
#pragma unroll
            for (int half = 0; half < 2; ++half) {
#pragma unroll
                for (int i = 0; i < 8; ++i) {
                    myLds[(half * 16 + mBase + i) * LDS_STRIDE + n] = acc[half][i];
                }
            }
        }
        asm volatile("s_wait_dscnt 0x0" ::: "memory");   // same-wave LDS RAW fence

        // ---- each lane: one full row -> 3-level tree reduce ----
        {
            const float* rp = myLds + lane * LDS_STRIDE;
            float h8[8];
#pragma unroll
            for (int i = 0; i < 8; ++i) h8[i] = rp[i];

            float g4[4];
#pragma unroll
            for (int i = 0; i < 4; ++i) {
                const float l = h8[2*i], r = h8[2*i+1], s = l + r;
                g4[i] = cA[i]*s + cB[i]*(l*r) + cC[i]*__sinf(s) + cD[i];
            }
            float g2[2];
#pragma unroll
            for (int i = 0; i < 2; ++i) {
                const float l = g4[2*i], r = g4[2*i+1], s = l + r;
                g2[i] = cA[4+i]*s + cB[4+i]*(l*r) + cC[4+i]*__sinf(s) + cD[4+i];
            }
            const float l = g2[0], r = g2[1], s = l + r;
            const float res = cA[6]*s + cB[6]*(l*r) + cC[6]*__sinf(s) + cD[6];

            const long long orow = rowBase + lane;
            if (orow < (long long)N) out[orow] = res;
        }
        asm volatile("s_wait_dscnt 0x0" ::: "memory");   // protect slab before next overwrite
    }
}

extern "C" void kernel_launch(void* const* d_in, const int* in_sizes, int n_in,
                              void* d_out, int out_size, void* d_ws, size_t ws_size,
                              hipStream_t stream)
{
    (void)n_in; (void)d_ws; (void)ws_size; (void)out_size;
    const float* x      = (const float*)d_in[0];
    const float* W_leaf = (const float*)d_in[1];
    const float* b_leaf = (const float*)d_in[2];
    const float* w0     = (const float*)d_in[3];
    const float* w1     = (const float*)d_in[4];
    const float* w2     = (const float*)d_in[5];
    const float* b0     = (const float*)d_in[6];
    const float* b1     = (const float*)d_in[7];
    const float* b2     = (const float*)d_in[8];
    const float* om0    = (const float*)d_in[9];
    const float* om1    = (const float*)d_in[10];
    const float* om2    = (const float*)d_in[11];
    float* out = (float*)d_out;

    const int N = in_sizes[0] / 16;
    // 256 threads = 8 waves/block; each wave covers 32 rows per iteration.
    int blocks = (N + 255) / 256;
    if (blocks < 1) blocks = 1;
    binary_tree_rnn_kernel<<<blocks, 256, 0, stream>>>(
        x, W_leaf, b_leaf, w0, w1, w2, b0, b1, b2, om0, om1, om2, out, N);
}